// CachedGlmExperts_80968723464471
// MI455X (gfx1250) — compile-verified
//
#include <hip/hip_runtime.h>
#include <hip/hip_bf16.h>
#include <math.h>

// Problem constants (match reference)
#define T_  128
#define H_  2048
#define E_  64
#define I2_ 2816
#define I_  1408          // I2_/2

// LDS strides padded so that rows map to distinct bank groups (stride % 64 == 4)
#define XS_STRIDE 2052    // x tile row stride (floats)
#define AS_STRIDE 1412    // act tile row stride (floats)

#define KB 8              // k-steps (of 4) per software-pipelined block: K-block = 32

typedef float v2f __attribute__((ext_vector_type(2)));
typedef float v8f __attribute__((ext_vector_type(8)));

// ---------------------------------------------------------------------------
// Kernel 0: zero the output accumulator
// ---------------------------------------------------------------------------
__global__ void zero_out_kernel(float* __restrict__ out, int n) {
    int i = blockIdx.x * blockDim.x + threadIdx.x;
    if (i < n) out[i] = 0.0f;
}

// ---------------------------------------------------------------------------
// Kernel 1: routing — softmax -> top-k -> renormalize, writes gate matrix
// ---------------------------------------------------------------------------
__global__ void route_topk_kernel(const float* __restrict__ logits,
                                  const int* __restrict__ topk_p,
                                  float* __restrict__ gmat) {
    __shared__ float sl[E_];
    const int t = blockIdx.x;
    const int e = threadIdx.x;
    sl[e] = logits[t * E_ + e];
    __syncthreads();
    if (e == 0) {
        int k = *topk_p;
        if (k > E_) k = E_;
        float mx = sl[0];
        for (int i = 1; i < E_; ++i) mx = fmaxf(mx, sl[i]);
        float q[E_];
        float sum = 0.0f;
        for (int i = 0; i < E_; ++i) { q[i] = __expf(sl[i] - mx); sum += q[i]; }
        const float inv = __builtin_amdgcn_rcpf(sum);
        for (int i = 0; i < E_; ++i) q[i] *= inv;

        float row[E_];
        for (int i = 0; i < E_; ++i) row[i] = 0.0f;
        float s = 0.0f;
        for (int j = 0; j < k; ++j) {
            int best = 0; float bv = -1.0f;
            for (int i = 0; i < E_; ++i) if (q[i] > bv) { bv = q[i]; best = i; }
            row[best] = bv; s += bv; q[best] = -2.0f;
        }
        const float rs = 1.0f / s;   // one accurate divide per token
        for (int i = 0; i < E_; ++i) gmat[t * E_ + i] = row[i] * rs;
    }
}

// ---------------------------------------------------------------------------
// Kernel 2: deterministic per-expert compaction of (token, gate) lists
// ---------------------------------------------------------------------------
__global__ void compact_kernel(const float* __restrict__ gmat,
                               int* __restrict__ counts,
                               int* __restrict__ tok_list,
                               float* __restrict__ gate_list) {
    const int e = threadIdx.x;   // 64 threads, 1 block
    int cnt = 0;
    for (int t = 0; t < T_; ++t) {
        const float g = gmat[t * E_ + e];
        if (g != 0.0f) {
            tok_list[e * T_ + cnt]  = t;
            gate_list[e * T_ + cnt] = g;
            ++cnt;
        }
    }
    counts[e] = cnt;
}

// ---------------------------------------------------------------------------
// Kernel 3: fused expert compute.
// Grid: E * 8 blocks, 256 threads (8 waves); 16 tokens of one expert per block.
//   Phase 1: act[16,1408] = silu(x@w1_gate^T) * (x@w1_up^T)   (WMMA f32 16x16x4)
//   Phase 2: out[t,:]    += gate * act @ w2^T                  (WMMA f32 16x16x4)
// Weights streamed once, non-temporal; loads software-pipelined in blocks of
// K=32 (16 independent global_load_b64 in flight before the WMMAs consume).
// ---------------------------------------------------------------------------
__global__ __launch_bounds__(256)
void moe_expert_kernel(const float* __restrict__ x,
                       const float* __restrict__ w1,
                       const float* __restrict__ w2,
                       const int*   __restrict__ counts,
                       const int*   __restrict__ tok_list,
                       const float* __restrict__ gate_list,
                       float* __restrict__ out) {
    extern __shared__ float smem[];
    float* xs    = smem;                          // [16][XS_STRIDE]
    float* acts  = xs + 16 * XS_STRIDE;           // [16][AS_STRIDE]
    int*   toks  = (int*)(acts + 16 * AS_STRIDE); // [16]
    float* gates = (float*)(toks + 16);           // [16]

    const int e    = blockIdx.x >> 3;
    const int tile = blockIdx.x & 7;
    const int cnt  = counts[e];
    if (tile * 16 >= cnt) return;

    const int tid   = threadIdx.x;
    const int lane  = tid & 31;
    const int wave  = tid >> 5;
    const int r     = lane & 15;   // fragment row/col selector (M or N = lane%16)
    const int khalf = lane >> 4;   // K sub-slice: K = 2*khalf + v

    // token metadata for this tile (pad with token 0 / gate 0)
    if (tid < 16) {
        const int s = tile * 16 + tid;
        int tk = 0; float g = 0.0f;
        if (s < cnt) { tk = tok_list[e * T_ + s]; g = gate_list[e * T_ + s]; }
        toks[tid] = tk; gates[tid] = g;
    }
    __syncthreads();

    // Stage x tile [16][2048] into LDS (float4 vectorized, padded stride)
    for (int idx = tid; idx < 16 * (H_ / 4); idx += 256) {
        const int row  = idx / (H_ / 4);
        const int col4 = idx % (H_ / 4);
        const float4 v = ((const float4*)(x + (size_t)toks[row] * H_))[col4];
        *(float4*)(xs + row * XS_STRIDE + col4 * 4) = v;
    }
    __syncthreads();

    // ---------------- Phase 1: x @ w1^T, SiLU-and-mul -> acts in LDS --------
    const size_t w1_base = (size_t)e * I2_ * H_;
    for (int it = wave; it < I_ / 16; it += 8) {
        const int i0 = it * 16;
        v8f c1 = {0.f,0.f,0.f,0.f,0.f,0.f,0.f,0.f};
        v8f c2 = c1;
        const float* ap  = xs + r * XS_STRIDE + 2 * khalf;
        const float* b1p = w1 + w1_base + (size_t)(i0 + r) * H_ + 2 * khalf;
        const float* b2p = w1 + w1_base + (size_t)(i0 + I_ + r) * H_ + 2 * khalf;
        for (int k = 0; k < H_; k += 4 * KB) {
            v2f a[KB], b1[KB], b2[KB];
            #pragma unroll
            for (int u = 0; u < KB; ++u) {
                b1[u] = __builtin_nontemporal_load((const v2f*)(b1p + k + 4 * u));
                b2[u] = __builtin_nontemporal_load((const v2f*)(b2p + k + 4 * u));
            }
            #pragma unroll
            for (int u = 0; u < KB; ++u)
                a[u] = *(const v2f*)(ap + k + 4 * u);
            #pragma unroll
            for (int u = 0; u < KB; ++u) {
                c1 = __builtin_amdgcn_wmma_f32_16x16x4_f32(false, a[u], false, b1[u],
                                                           (short)0, c1, false, false);
                c2 = __builtin_amdgcn_wmma_f32_16x16x4_f32(false, a[u], false, b2[u],
                                                           (short)0, c2, false, false);
            }
        }
        // C layout: VGPR j -> M = j + 8*khalf, N = r
        const int col   = i0 + r;
        const int mbase = khalf * 8;
        #pragma unroll
        for (int j = 0; j < 8; ++j) {
            const float g   = c1[j];
            // silu(g) * up : g * sigmoid(g); single v_rcp_f32, no IEEE div chain
            const float sig = __builtin_amdgcn_rcpf(1.0f + __expf(-g));
            acts[(mbase + j) * AS_STRIDE + col] = g * sig * c2[j];
        }
    }
    __syncthreads();

    // ---------------- Phase 2: acts @ w2^T, gated atomic accumulate ---------
    const size_t w2_base = (size_t)e * H_ * I_;
    for (int ht = wave; ht < H_ / 16; ht += 8) {
        const int h0 = ht * 16;
        v8f c = {0.f,0.f,0.f,0.f,0.f,0.f,0.f,0.f};
        const float* ap = acts + r * AS_STRIDE + 2 * khalf;
        const float* bp = w2 + w2_base + (size_t)(h0 + r) * I_ + 2 * khalf;
        for (int k = 0; k < I_; k += 4 * KB) {
            v2f a[KB], b[KB];
            #pragma unroll
            for (int u = 0; u < KB; ++u)
                b[u] = __builtin_nontemporal_load((const v2f*)(bp + k + 4 * u));
            #pragma unroll
            for (int u = 0; u < KB; ++u)
                a[u] = *(const v2f*)(ap + k + 4 * u);
            #pragma unroll
            for (int u = 0; u < KB; ++u)
                c = __builtin_amdgcn_wmma_f32_16x16x4_f32(false, a[u], false, b[u],
                                                          (short)0, c, false, false);
        }
        const int n     = h0 + r;
        const int mbase = khalf * 8;
        #pragma unroll
        for (int j = 0; j < 8; ++j) {
            const int m = mbase + j;
            const float v = gates[m] * c[j];
            unsafeAtomicAdd(out + (size_t)toks[m] * H_ + n, v);
        }
    }
}

// ---------------------------------------------------------------------------
// Host entry
// ---------------------------------------------------------------------------
extern "C" void kernel_launch(void* const* d_in, const int* in_sizes, int n_in,
                              void* d_out, int out_size, void* d_ws, size_t ws_size,
                              hipStream_t stream) {
    const float* x      = (const float*)d_in[0];   // [T,H]
    const float* logits = (const float*)d_in[1];   // [T,E]
    const float* w1     = (const float*)d_in[2];   // [E,2I,H]
    const float* w2     = (const float*)d_in[3];   // [E,H,I]
    const int*   topk   = (const int*)d_in[4];     // scalar
    float* out = (float*)d_out;                    // [T,1,H] = T*H floats

    // Workspace layout (256B-aligned chunks)
    char* ws = (char*)d_ws;
    float* gmat      = (float*)(ws);                       // T*E floats  (32 KB)
    int*   counts    = (int*)  (ws + 33024);               // E ints
    int*   tok_list  = (int*)  (ws + 33536);               // E*T ints    (32 KB)
    float* gate_list = (float*)(ws + 66560);               // E*T floats  (32 KB)

    const int n_out = T_ * H_;
    zero_out_kernel<<<(n_out + 255) / 256, 256, 0, stream>>>(out, n_out);
    route_topk_kernel<<<T_, E_, 0, stream>>>(logits, topk, gmat);
    compact_kernel<<<1, E_, 0, stream>>>(gmat, counts, tok_list, gate_list);

    const size_t smem_bytes =
        (size_t)(16 * XS_STRIDE + 16 * AS_STRIDE) * sizeof(float)
        + 16 * sizeof(int) + 16 * sizeof(float);
    static bool attr_set = false;
    if (!attr_set) {
        (void)hipFuncSetAttribute((const void*)moe_expert_kernel,
                                  hipFuncAttributeMaxDynamicSharedMemorySize,
                                  (int)smem_bytes);
        attr_set = true;
    }
    moe_expert_kernel<<<E_ * 8, 256, smem_bytes, stream>>>(
        x, w1, w2, counts, tok_list, gate_list, out);
}